// AffineTransformLayer_90168543412829
// MI455X (gfx1250) — compile-verified
//
#include <hip/hip_runtime.h>
#include <hip/hip_bf16.h>
#include <stdint.h>

// Dims fixed by the reference: x:[8,4,1024,1024] f32, transform:[8,6] f32.
#define AT_B 8
#define AT_C 4
#define AT_H 1024
#define AT_W 1024
#define AT_PSTRIDE 16   // floats per batch in the param workspace

typedef float v4f __attribute__((ext_vector_type(4)));

typedef __attribute__((address_space(1))) int g_i32;   // global-AS int
typedef __attribute__((address_space(3))) int l_i32;   // LDS-AS int

// Probe the CDNA5 async Global->LDS DMA path (ASYNCcnt-tracked).
#if __has_builtin(__builtin_amdgcn_global_load_async_to_lds_b32) && \
    __has_builtin(__builtin_amdgcn_s_wait_asynccnt)
#define AT_ASYNC_LDS 1
#endif

// ---------------------------------------------------------------------------
// Setup: per-batch 2x2 inverse, inverse translation, scalar bilinear weights.
// ws[b*16 + {0..9}] = inv00, inv01, inv10, inv11, ti0, ti1, w00, w10, w01, w11
// ---------------------------------------------------------------------------
__global__ void at_setup_params(const float* __restrict__ tr,
                                float* __restrict__ ws) {
  int b = (int)threadIdx.x;
  if (b >= AT_B) return;
  const float* t = tr + b * 6;
  float a00 = t[0], a01 = t[1], a10 = t[2], a11 = t[3];
  float t0 = t[4], t1 = t[5];

  float det   = a00 * a11 - a01 * a10;
  float inv00 =  a11 / det;
  float inv01 = -a01 / det;
  float inv10 = -a10 / det;
  float inv11 =  a00 / det;

  // t_inv = -(t @ Ainv)
  float ti0 = -(t0 * inv00 + t1 * inv10);
  float ti1 = -(t0 * inv01 + t1 * inv11);

  // Fractional offsets of pixel 0 (the reference's per-batch scalar weights).
  float c0 = fminf(fmaxf(ti0, 0.0f), (float)(AT_H - 2));
  float c1 = fminf(fmaxf(ti1, 0.0f), (float)(AT_W - 2));
  float dx = c0 - (float)(int)c0;
  float dy = c1 - (float)(int)c1;

  float* p = ws + (size_t)b * AT_PSTRIDE;
  p[0] = inv00; p[1] = inv01; p[2] = inv10; p[3] = inv11;
  p[4] = ti0;   p[5] = ti1;
  p[6] = (1.0f - dx) * (1.0f - dy);   // w00 : (y0, x0)
  p[7] = dx * (1.0f - dy);            // w10 : (y0, x0+1)
  p[8] = (1.0f - dx) * dy;            // w01 : (y0+1, x0)
  p[9] = dx * dy;                     // w11 : (y0+1, x0+1)
  p[10] = 0.0f; p[11] = 0.0f;         // pad (async stage loads 12 dwords)
}

// ---------------------------------------------------------------------------
// Main: one block per output row of one (b,c) image. 256 thr x 4 px = 1024.
// Gathers stay regular-temporal (source fits in 192MB L2); output is streamed
// with a non-temporal b128 store so it never evicts the resident source.
// ---------------------------------------------------------------------------
__global__ __launch_bounds__(256) void at_affine_gather(
    const float* __restrict__ img, const float* __restrict__ ws,
    float* __restrict__ out) {
  unsigned lin = blockIdx.x;          // [0, B*C*H)
  unsigned y   = lin & (AT_H - 1);
  unsigned bc  = lin >> 10;           // b*C + c
  unsigned b   = bc >> 2;

  __shared__ float p[AT_PSTRIDE];
  const float* gp = ws + (size_t)b * AT_PSTRIDE;

#if defined(AT_ASYNC_LDS)
  // Stage the 12-dword param block into LDS via the async DMA path
  // (global_load_async_to_lds_b32, tracked by ASYNCcnt).
  if (threadIdx.x < 12) {
    g_i32* ga = (g_i32*)(uintptr_t)(gp + threadIdx.x);
    l_i32* la = (l_i32*)(uint32_t)(uintptr_t)(p + threadIdx.x);
    __builtin_amdgcn_global_load_async_to_lds_b32(ga, la, 0, 0);
  }
  __builtin_amdgcn_s_wait_asynccnt(0);
  __syncthreads();
#else
  if (threadIdx.x < 12) p[threadIdx.x] = gp[threadIdx.x];
  __syncthreads();
#endif

  float ai00 = p[0], ai01 = p[1], ai10 = p[2], ai11 = p[3];
  float ti0  = p[4], ti1  = p[5];
  float w00  = p[6], w10  = p[7], w01 = p[8], w11 = p[9];

  float fy = (float)y;
  const float* src = img + ((size_t)bc << 20);                 // H*W = 1<<20
  float*       dst = out + ((size_t)bc << 20) + ((size_t)y << 10);

  unsigned x0 = threadIdx.x * 4u;
  v4f r;
#pragma unroll
  for (int i = 0; i < 4; ++i) {
    float fx = (float)(x0 + (unsigned)i);
    // Same association as the reference: (x*inv + y*inv) + t_inv
    float u = fx * ai00 + fy * ai10 + ti0;
    float v = fx * ai01 + fy * ai11 + ti1;
    u = fminf(fmaxf(u, 0.0f), (float)(AT_H - 2));
    v = fminf(fmaxf(v, 0.0f), (float)(AT_W - 2));
    int i0 = (int)u;  // source column
    int i1 = (int)v;  // source row
    const float* s = src + ((size_t)(unsigned)i1 << 10) + (unsigned)i0;
    float g00 = s[0];
    float g10 = s[1];
    float g01 = s[AT_W];
    float g11 = s[AT_W + 1];
    r[i] = w00 * g00 + w10 * g10 + w01 * g01 + w11 * g11;
  }
  __builtin_nontemporal_store(r, (v4f*)(dst + x0));
}

// ---------------------------------------------------------------------------
extern "C" void kernel_launch(void* const* d_in, const int* in_sizes, int n_in,
                              void* d_out, int out_size, void* d_ws, size_t ws_size,
                              hipStream_t stream) {
  (void)in_sizes; (void)n_in; (void)out_size; (void)ws_size;
  const float* x  = (const float*)d_in[0];   // [8,4,1024,1024] f32
  const float* tr = (const float*)d_in[1];   // [8,6] f32
  float* out = (float*)d_out;                // [8,4,1024,1024] f32
  float* ws  = (float*)d_ws;                 // >= 8*16 floats

  at_setup_params<<<1, 32, 0, stream>>>(tr, ws);
  at_affine_gather<<<AT_B * AT_C * AT_H, 256, 0, stream>>>(x, ws, out);
}